// Interp_String_69741678953241
// MI455X (gfx1250) — compile-verified
//
#include <hip/hip_runtime.h>
#include <stdint.h>

#define N       8192
#define NW      256        // N/32 mask words per row
#define SCALE   0.05333333f
#define R1SQ    2.25f      // 1.5^2
#define R2SQ    1.44f      // 1.2^2
#define LOSQ    0.9025f    // 0.95^2
#define HISQ    1.1025f    // 1.05^2
#define COINC   0.90f      // threshold on squared distance (per reference)
#define REFILL  409        // int(N*0.05)
#define NITERS  8

typedef float v2f __attribute__((ext_vector_type(2)));
typedef float v8f __attribute__((ext_vector_type(8)));
typedef unsigned int uint;

__device__ __forceinline__ uint h32(uint x) {
    x ^= x >> 16; x *= 0x7feb352du;
    x ^= x >> 15; x *= 0x846ca68bu;
    x ^= x >> 16; return x;
}

// ---------------------------------------------------------------------------
// Kernel 0: scale coords, squared norms, zero pinned state
// ---------------------------------------------------------------------------
__global__ __launch_bounds__(256) void prepKernel(const float* __restrict__ pts,
                                                  float* __restrict__ xs,
                                                  float* __restrict__ ys,
                                                  float* __restrict__ sq,
                                                  int* __restrict__ pinned) {
    int i = blockIdx.x * 256 + threadIdx.x;
    if (i < N) {
        float x = pts[i * 5 + 0] * SCALE;
        float y = pts[i * 5 + 1] * SCALE;
        xs[i] = x; ys[i] = y; sq[i] = x * x + y * y;
        pinned[i] = 0;
    }
}

// ---------------------------------------------------------------------------
// Tile helper: one 16x16 d^2 tile via V_WMMA_F32_16X16X4_F32 (K=2 padded).
// C layout: lane holds col N = lane&15; VGPR r holds row r (lanes 0-15)
// and row r+8 (lanes 16-31).  Ballot packs two rows' 16-bit masks per word.
// B operand loaded unconditionally (ds_load_b64), K-pad lanes zeroed by a
// hoisted 0/1 multiplier -> no exec save/restore in the inner loop.
// ---------------------------------------------------------------------------
__device__ __forceinline__ void tile_ballots(
    v2f av, int lane, float kmask,
    const v2f* __restrict__ lxy,
    const float* __restrict__ lsq, const int* __restrict__ limg,
    int tOff /*elem offset in chunk*/, int col0 /*global col base*/,
    const float sqi[8], const int imgi[8], const int rowIdx[8],
    uint adj[8], uint coin[8], uint pas[8], uint fal[8]) {
    int jl = tOff + (lane & 15);
    v2f bxy = lxy[jl];                 // one ds_load_b64, all lanes
    v2f bv;
    bv.x = bxy.x * kmask;              // lanes 16-31 -> 0 (K=2,3 zero pad)
    bv.y = bxy.y * kmask;
    float sqj = lsq[jl];
    int   imgj = limg[jl];
    int   jcol = col0 + (lane & 15);

    v8f c = {};
    c = __builtin_amdgcn_wmma_f32_16x16x4_f32(
            /*neg_a=*/false, av, /*neg_b=*/false, bv,
            /*c_mod=*/(short)0, c, /*reuse_a=*/false, /*reuse_b=*/false);

#pragma unroll
    for (int r = 0; r < 8; ++r) {
        float d2 = sqi[r] + sqj - 2.0f * c[r];
        d2 = fmaxf(d2, 1e-12f);
        bool same = (imgi[r] == imgj);
        bool diag = (rowIdx[r] == jcol);
        bool base = same && !diag;
        bool a1 = base && (d2 < R1SQ);
        bool r2 = base && (d2 < R2SQ);
        bool co = base && (d2 < COINC);
        bool inband = (d2 > LOSQ) && (d2 < HISQ);
        bool pa = r2 && inband;
        bool fa = r2 && !inband;
        adj[r]  = __builtin_amdgcn_ballot_w32(a1);
        coin[r] = __builtin_amdgcn_ballot_w32(co);
        pas[r]  = __builtin_amdgcn_ballot_w32(pa);
        fal[r]  = __builtin_amdgcn_ballot_w32(fa);
    }
}

// ---------------------------------------------------------------------------
// Kernel 1: single O(N^2) sweep.  Each wave owns 16 rows, loops all columns
// (staged through LDS in 2048-point chunks), emits deg[] plus four N x N
// bitmasks (32 MB total -> L2 resident on MI455X's 192 MB L2).
// ---------------------------------------------------------------------------
__global__ __launch_bounds__(256) void pairPassA(
    const float* __restrict__ xs, const float* __restrict__ ys,
    const float* __restrict__ sq, const int* __restrict__ imgid,
    int* __restrict__ deg,
    uint* __restrict__ adjm, uint* __restrict__ coinm,
    uint* __restrict__ pasm, uint* __restrict__ falm) {
    __shared__ v2f   lxy[2048];
    __shared__ float lsq[2048];
    __shared__ int   limg[2048];

    int lane = threadIdx.x & 31;
    int wave = threadIdx.x >> 5;
    int row0 = blockIdx.x * 128 + wave * 16;
    float kmask = (lane < 16) ? 1.0f : 0.0f;

    // A operand: lanes 0-15 hold (x,y) of rows row0..row0+15; lanes 16-31 zero (K pad)
    v2f av;
    {
        int ar = row0 + (lane & 15);
        av.x = xs[ar] * kmask;
        av.y = ys[ar] * kmask;
    }
    float sqi[8]; int imgi[8], rowIdx[8];
    int rbase = row0 + ((lane >> 4) << 3);     // lanes>=16 own rows +8
#pragma unroll
    for (int r = 0; r < 8; ++r) {
        int rr = rbase + r;
        sqi[r] = sq[rr]; imgi[r] = imgid[rr]; rowIdx[r] = rr;
    }
    int degacc[8];
#pragma unroll
    for (int r = 0; r < 8; ++r) degacc[r] = 0;

    for (int ch = 0; ch < 4; ++ch) {
        int base = ch * 2048;
        __syncthreads();
        for (int k = threadIdx.x; k < 2048; k += 256) {
            v2f t; t.x = xs[base + k]; t.y = ys[base + k];
            lxy[k] = t;
            lsq[k] = sq[base + k]; limg[k] = imgid[base + k];
        }
        __syncthreads();

        for (int tp = 0; tp < 64; ++tp) {          // 64 word-pairs = 128 tiles / chunk
            int tOffE = tp * 32;
            int col0E = base + tOffE;
            uint aE[8], cE[8], pE[8], fE[8];
            uint aO[8], cO[8], pO[8], fO[8];
            tile_ballots(av, lane, kmask, lxy, lsq, limg, tOffE,      col0E,
                         sqi, imgi, rowIdx, aE, cE, pE, fE);
            tile_ballots(av, lane, kmask, lxy, lsq, limg, tOffE + 16, col0E + 16,
                         sqi, imgi, rowIdx, aO, cO, pO, fO);
            int w = (base >> 5) + tp;
            bool writer = (lane == 0) || (lane == 16);
#pragma unroll
            for (int r = 0; r < 8; ++r) {
                bool lo = (lane < 16);
                uint wAdj = lo ? ((aE[r] & 0xffffu) | ((aO[r] & 0xffffu) << 16))
                               : ((aE[r] >> 16)     | (aO[r] & 0xffff0000u));
                uint wCoi = lo ? ((cE[r] & 0xffffu) | ((cO[r] & 0xffffu) << 16))
                               : ((cE[r] >> 16)     | (cO[r] & 0xffff0000u));
                uint wPas = lo ? ((pE[r] & 0xffffu) | ((pO[r] & 0xffffu) << 16))
                               : ((pE[r] >> 16)     | (pO[r] & 0xffff0000u));
                uint wFal = lo ? ((fE[r] & 0xffffu) | ((fO[r] & 0xffffu) << 16))
                               : ((fE[r] >> 16)     | (fO[r] & 0xffff0000u));
                degacc[r] += __popc(wAdj);
                if (writer) {
                    size_t o = (size_t)rowIdx[r] * NW + (size_t)w;
                    adjm[o] = wAdj; coinm[o] = wCoi; pasm[o] = wPas; falm[o] = wFal;
                }
            }
        }
    }
    if (lane == 0 || lane == 16) {
#pragma unroll
        for (int r = 0; r < 8; ++r) deg[rowIdx[r]] = degacc[r];
    }
}

// ---------------------------------------------------------------------------
// Kernel 2: end points — no adj1 neighbor with smaller degree
// ---------------------------------------------------------------------------
__global__ __launch_bounds__(256) void endPointKernel(
    const uint* __restrict__ adjm, const int* __restrict__ deg,
    uint* __restrict__ end_words) {
    int i = blockIdx.x * 256 + threadIdx.x;
    int di = deg[i];
    bool isEnd = true;
    const uint* row = adjm + (size_t)i * NW;
    for (int w = 0; w < NW && isEnd; ++w) {
        uint m = row[w];
        while (m) {
            int b = __ffs(m) - 1; m &= m - 1;
            if (di > deg[w * 32 + b]) { isEnd = false; break; }
        }
    }
    uint bal = __builtin_amdgcn_ballot_w32(isEnd);
    if ((threadIdx.x & 31) == 0) end_words[i >> 5] = bal;
}

// ---------------------------------------------------------------------------
// Kernel 3: drop coincident left endpoints (triangular j>i), final end set,
// and hash-based 5% random + end_final initial active mask
// ---------------------------------------------------------------------------
__global__ __launch_bounds__(256) void endFinalKernel(
    const uint* __restrict__ coinm, const uint* __restrict__ end_words,
    uint* __restrict__ endfin_words, uint* __restrict__ act0) {
    int i = blockIdx.x * 256 + threadIdx.x;
    bool endi = (end_words[i >> 5] >> (i & 31)) & 1u;
    bool dropped = false;
    if (endi) {
        const uint* row = coinm + (size_t)i * NW;
        int w0 = i >> 5;
        for (int w = w0; w < NW && !dropped; ++w) {
            uint m = row[w] & end_words[w];
            if (w == w0) {
                int b = i & 31;
                m &= (b == 31) ? 0u : (~0u << (b + 1));   // strictly j > i
            }
            if (m) dropped = true;
        }
    }
    bool ef  = endi && !dropped;
    bool act = ef || (h32((uint)i * 2654435761u + 12345u) < 214748364u); // ~5%
    uint b1 = __builtin_amdgcn_ballot_w32(ef);
    uint b2 = __builtin_amdgcn_ballot_w32(act);
    if ((threadIdx.x & 31) == 0) {
        endfin_words[i >> 5] = b1;
        act0[i >> 5] = b2;
    }
}

// ---------------------------------------------------------------------------
// Kernel 4: deterministic refill mask (409 hashed indices) per iteration
// ---------------------------------------------------------------------------
__global__ __launch_bounds__(256) void refillKernel(uint* __restrict__ refill_words,
                                                    int iter) {
    int w = threadIdx.x;
    uint m = 0;
    for (int k = 0; k < REFILL; ++k) {
        uint idx = h32((uint)(iter + 1) * 100003u + (uint)k * 7919u) & (N - 1);
        if ((int)(idx >> 5) == w) m |= 1u << (idx & 31);
    }
    refill_words[w] = m;
}

// ---------------------------------------------------------------------------
// Kernel 5: one pinning iteration — bit-AND row scans against the LDS-staged
// 8192-bit active mask (mask bitmaps stay in L2)
// ---------------------------------------------------------------------------
__global__ __launch_bounds__(256) void pinIterKernel(
    const uint* __restrict__ pasm, const uint* __restrict__ falm,
    const uint* __restrict__ actIn, uint* __restrict__ actOut,
    const uint* __restrict__ endfin_words, const uint* __restrict__ refill_words,
    int* __restrict__ pinned) {
    __shared__ uint aw[NW];
    aw[threadIdx.x] = actIn[threadIdx.x];
    __syncthreads();
    int i = blockIdx.x * 256 + threadIdx.x;
    bool acti = (aw[i >> 5] >> (i & 31)) & 1u;
    uint fAny = 0, pAny = 0;
    if (acti) {
        const uint* fr = falm + (size_t)i * NW;
        const uint* pr = pasm + (size_t)i * NW;
        for (int w = 0; w < NW; ++w) {
            uint u = aw[w];
            fAny |= fr[w] & u;
            pAny |= pr[w] & u;
        }
    }
    bool fail_n = (fAny != 0);
    bool pass_n = (pAny != 0) && !fail_n;
    int pin = pinned[i] | (pass_n ? 1 : 0);
    pinned[i] = pin;
    bool ef = (endfin_words[i >> 5] >> (i & 31)) & 1u;
    bool rf = (refill_words[i >> 5] >> (i & 31)) & 1u;
    bool na = (acti && !fail_n) || (pin != 0) || ef || rf;
    uint b = __builtin_amdgcn_ballot_w32(na);
    if ((threadIdx.x & 31) == 0) actOut[i >> 5] = b;
}

// ---------------------------------------------------------------------------
// Kernel 6: out = pts * active
// ---------------------------------------------------------------------------
__global__ __launch_bounds__(256) void maskOutKernel(const float* __restrict__ pts,
                                                     const uint* __restrict__ actWords,
                                                     float* __restrict__ out) {
    int k = blockIdx.x * 256 + threadIdx.x;
    if (k < N * 5) {
        int i = k / 5;
        float m = ((actWords[i >> 5] >> (i & 31)) & 1u) ? 1.0f : 0.0f;
        out[k] = pts[k] * m;
    }
}

// ---------------------------------------------------------------------------
extern "C" void kernel_launch(void* const* d_in, const int* in_sizes, int n_in,
                              void* d_out, int out_size, void* d_ws, size_t ws_size,
                              hipStream_t stream) {
    (void)in_sizes; (void)n_in; (void)out_size; (void)ws_size;
    const float* pts   = (const float*)d_in[0];   // [N,5] f32
    const int*   imgid = (const int*)d_in[1];     // [N]  i32
    float* out = (float*)d_out;

    char* p = (char*)d_ws;
    float* xs  = (float*)p; p += (size_t)N * 4;
    float* ys  = (float*)p; p += (size_t)N * 4;
    float* sq  = (float*)p; p += (size_t)N * 4;
    int*   deg = (int*)p;   p += (size_t)N * 4;
    int*   pin = (int*)p;   p += (size_t)N * 4;
    uint* adjm  = (uint*)p; p += (size_t)N * NW * 4;   // 8 MB each
    uint* coinm = (uint*)p; p += (size_t)N * NW * 4;
    uint* pasm  = (uint*)p; p += (size_t)N * NW * 4;
    uint* falm  = (uint*)p; p += (size_t)N * NW * 4;
    uint* end_words    = (uint*)p; p += NW * 4;
    uint* endfin_words = (uint*)p; p += NW * 4;
    uint* actA = (uint*)p; p += NW * 4;
    uint* actB = (uint*)p; p += NW * 4;
    uint* refill_words = (uint*)p; p += NW * 4;

    prepKernel<<<N / 256, 256, 0, stream>>>(pts, xs, ys, sq, pin);
    pairPassA<<<N / 128, 256, 0, stream>>>(xs, ys, sq, imgid, deg,
                                           adjm, coinm, pasm, falm);
    endPointKernel<<<N / 256, 256, 0, stream>>>(adjm, deg, end_words);
    endFinalKernel<<<N / 256, 256, 0, stream>>>(coinm, end_words, endfin_words, actA);

    uint* bufs[2] = { actA, actB };
    for (int it = 0; it < NITERS; ++it) {
        refillKernel<<<1, 256, 0, stream>>>(refill_words, it);
        pinIterKernel<<<N / 256, 256, 0, stream>>>(pasm, falm,
                                                   bufs[it & 1], bufs[(it + 1) & 1],
                                                   endfin_words, refill_words, pin);
    }
    // after 8 iterations the final mask is back in bufs[0] (actA)
    maskOutKernel<<<(N * 5 + 255) / 256, 256, 0, stream>>>(pts, actA, out);
}